// GCNLayer_60842506715384
// MI455X (gfx1250) — compile-verified
//
#include <hip/hip_runtime.h>

typedef __attribute__((ext_vector_type(2))) float v2f;
typedef __attribute__((ext_vector_type(8))) float v8f;

// ---------------------------------------------------------------------------
// Degree init: deg[i] = 1.0 (self loop); also zero pool sums and counts.
// ---------------------------------------------------------------------------
__global__ __launch_bounds__(256) void k_init(float* __restrict__ deg,
                                              float* __restrict__ pool,
                                              float* __restrict__ cnt, int N) {
  int t = blockIdx.x * 256 + threadIdx.x;
  if (t < N) deg[t] = 1.0f;
  if (t < 64 * 128) pool[t] = 0.0f;
  if (t < 64) cnt[t] = 0.0f;
}

__global__ __launch_bounds__(256) void k_deg(const int* __restrict__ dst,
                                             float* __restrict__ deg, int E) {
  int e = blockIdx.x * 256 + threadIdx.x;
  if (e < E) atomicAdd(&deg[dst[e]], 1.0f);
}

__global__ __launch_bounds__(256) void k_rsqrt(float* __restrict__ deg, int N) {
  int i = blockIdx.x * 256 + threadIdx.x;
  if (i < N) deg[i] = rsqrtf(deg[i]);  // deg >= 1 always (self loops)
}

// ---------------------------------------------------------------------------
// Y[M,128] = X[M,128] @ W[128,128] via V_WMMA_F32_16X16X4_F32.
// Block = 256 threads = 8 waves; block covers 16 rows x 128 cols,
// wave w owns the 16x16 tile at column w*16. K-loop: 128/4 = 32 WMMAs.
// A layout (16x4 f32): lane%16 = M row; lane/16 selects K pair {0,1}/{2,3}.
// B layout (4x16):     lane%16 = N col; VGPR0/1 hold K=klo / klo+1.
// C/D (16x16 f32): 8 VGPRs, n = lane%16, m = v + 8*(lane/16).
// ---------------------------------------------------------------------------
__global__ __launch_bounds__(256) void k_gemm128(const float* __restrict__ X,
                                                 const float* __restrict__ W,
                                                 float* __restrict__ Y, int M) {
  const int lane = threadIdx.x & 31;
  const int wave = threadIdx.x >> 5;
  const int m0   = blockIdx.x * 16;
  const int mr   = lane & 15;
  const int klo  = (lane >> 4) * 2;
  const int n    = wave * 16 + mr;

  int arow = m0 + mr;
  if (arow >= M) arow = M - 1;  // clamp (M is a multiple of 16 here anyway)
  const float* __restrict__ xrow = X + (size_t)arow * 128;

  v8f c = {};
  for (int k0 = 0; k0 < 128; k0 += 4) {
    v2f a = *(const v2f*)(xrow + k0 + klo);          // 64-bit load
    v2f b;
    b.x = W[(size_t)(k0 + klo) * 128 + n];
    b.y = W[(size_t)(k0 + klo + 1) * 128 + n];
    c = __builtin_amdgcn_wmma_f32_16x16x4_f32(false, a, false, b,
                                              (short)0, c, false, false);
  }
#pragma unroll
  for (int v = 0; v < 8; ++v) {
    int m = m0 + v + 8 * (lane >> 4);
    if (m < M) Y[(size_t)m * 128 + n] = c[v];
  }
}

// ---------------------------------------------------------------------------
// agg[i] = xw[i] * dinv[i]^2   (self-loop contribution; also inits agg).
// One float4 per thread, 32 threads per node.
// ---------------------------------------------------------------------------
__global__ __launch_bounds__(256) void k_selfloop(const float* __restrict__ xw,
                                                  const float* __restrict__ dinv,
                                                  float* __restrict__ agg, int N) {
  int t = blockIdx.x * 256 + threadIdx.x;
  int node = t >> 5, q = t & 31;
  if (node >= N) return;
  float s = dinv[node];
  s = s * s;
  float4 v = ((const float4*)(xw + (size_t)node * 128))[q];
  float4 o = make_float4(v.x * s, v.y * s, v.z * s, v.w * s);
  ((float4*)(agg + (size_t)node * 128))[q] = o;
}

// ---------------------------------------------------------------------------
// Edge scatter: one wave per edge. Gather xw[src] (b128 per lane), scale by
// dinv[src]*dinv[dst], atomic-add into agg[dst]. L2 (192MB) holds the whole
// 51MB xw buffer, so random gathers mostly hit L2.
// ---------------------------------------------------------------------------
__global__ __launch_bounds__(256) void k_edge(const float* __restrict__ xw,
                                              float* __restrict__ agg,
                                              const int* __restrict__ src,
                                              const int* __restrict__ dst,
                                              const float* __restrict__ dinv, int E) {
  int e = (blockIdx.x * 256 + threadIdx.x) >> 5;
  int lane = threadIdx.x & 31;
  if (e >= E) return;
  int s = src[e], d = dst[e];
  float norm = dinv[s] * dinv[d];
  float4 v = ((const float4*)(xw + (size_t)s * 128))[lane];
  float* a = agg + (size_t)d * 128 + lane * 4;
  atomicAdd(a + 0, v.x * norm);
  atomicAdd(a + 1, v.y * norm);
  atomicAdd(a + 2, v.z * norm);
  atomicAdd(a + 3, v.w * norm);
}

// ---------------------------------------------------------------------------
// h = relu(agg + bias), in place.
// ---------------------------------------------------------------------------
__global__ __launch_bounds__(256) void k_bias_relu(float* __restrict__ h,
                                                   const float* __restrict__ bias,
                                                   int N) {
  int t = blockIdx.x * 256 + threadIdx.x;
  int node = t >> 5, q = t & 31;
  if (node >= N) return;
  float4 b = ((const float4*)bias)[q];
  float4 v = ((float4*)(h + (size_t)node * 128))[q];
  v.x = fmaxf(v.x + b.x, 0.0f);
  v.y = fmaxf(v.y + b.y, 0.0f);
  v.z = fmaxf(v.z + b.z, 0.0f);
  v.w = fmaxf(v.w + b.w, 0.0f);
  ((float4*)(h + (size_t)node * 128))[q] = v;
}

// ---------------------------------------------------------------------------
// Mean-pool accumulation: one wave per node, atomic-add into pool[batch[i]].
// ---------------------------------------------------------------------------
__global__ __launch_bounds__(256) void k_pool(const float* __restrict__ h,
                                              const int* __restrict__ batch,
                                              float* __restrict__ pool,
                                              float* __restrict__ cnt, int N) {
  int node = (blockIdx.x * 256 + threadIdx.x) >> 5;
  int lane = threadIdx.x & 31;
  if (node >= N) return;
  int g = batch[node];
  float4 v = ((const float4*)(h + (size_t)node * 128))[lane];
  float* p = pool + (size_t)g * 128 + lane * 4;
  atomicAdd(p + 0, v.x);
  atomicAdd(p + 1, v.y);
  atomicAdd(p + 2, v.z);
  atomicAdd(p + 3, v.w);
  if (lane == 0) atomicAdd(&cnt[g], 1.0f);
}

// ---------------------------------------------------------------------------
// Head: out[64,64] = (pool/cnt) @ Wl[128,64] + bl. Single block, 16 waves,
// each wave one 16x16 tile, WMMA f32 16x16x4, mean + bias fused.
// ---------------------------------------------------------------------------
__global__ __launch_bounds__(512) void k_final(const float* __restrict__ pool,
                                               const float* __restrict__ cnt,
                                               const float* __restrict__ Wl,
                                               const float* __restrict__ bl,
                                               float* __restrict__ out) {
  const int lane = threadIdx.x & 31;
  const int wave = threadIdx.x >> 5;
  const int m0 = (wave >> 2) * 16;
  const int n0 = (wave & 3) * 16;
  const int mr = lane & 15;
  const int klo = (lane >> 4) * 2;
  const int n = n0 + mr;

  float rc = 1.0f / fmaxf(cnt[m0 + mr], 1.0f);
  const float* __restrict__ prow = pool + (size_t)(m0 + mr) * 128;

  v8f c = {};
  for (int k0 = 0; k0 < 128; k0 += 4) {
    v2f a = *(const v2f*)(prow + k0 + klo);
    a.x *= rc;
    a.y *= rc;
    v2f b;
    b.x = Wl[(size_t)(k0 + klo) * 64 + n];
    b.y = Wl[(size_t)(k0 + klo + 1) * 64 + n];
    c = __builtin_amdgcn_wmma_f32_16x16x4_f32(false, a, false, b,
                                              (short)0, c, false, false);
  }
  float bias = bl[n];
#pragma unroll
  for (int v = 0; v < 8; ++v) {
    int m = m0 + v + 8 * (lane >> 4);
    out[(size_t)m * 64 + n] = c[v] + bias;
  }
}

// ---------------------------------------------------------------------------
extern "C" void kernel_launch(void* const* d_in, const int* in_sizes, int n_in,
                              void* d_out, int out_size, void* d_ws, size_t ws_size,
                              hipStream_t stream) {
  const float* x  = (const float*)d_in[0];  // [N,128]
  const int*   ei = (const int*)d_in[1];    // [2,E]
  const int*   bt = (const int*)d_in[2];    // [N]
  const float* W1 = (const float*)d_in[3];
  const float* b1 = (const float*)d_in[4];
  const float* W2 = (const float*)d_in[5];
  const float* b2 = (const float*)d_in[6];
  const float* Wl = (const float*)d_in[7];
  const float* bl = (const float*)d_in[8];
  float* out = (float*)d_out;

  const int N = in_sizes[2];
  const int E = in_sizes[1] / 2;
  const int* src = ei;
  const int* dst = ei + E;

  // Workspace layout (floats)
  float* ws   = (float*)d_ws;
  float* bufA = ws;                           // [N,128]  xw
  float* bufB = bufA + (size_t)N * 128;       // [N,128]  agg / h
  float* dinv = bufB + (size_t)N * 128;       // [N]      deg -> dinv
  float* pool = dinv + N;                     // [64,128]
  float* cnt  = pool + 64 * 128;              // [64]

  const int nBlkN   = (N + 255) / 256;
  const int nBlkE   = (E + 255) / 256;
  const int nBlkN32 = ((size_t)N * 32 + 255) / 256;  // 1 wave per node
  const int nBlkEw  = (E + 7) / 8;                   // 1 wave per edge
  const int nBlkNw  = (N + 7) / 8;
  const int gemmBlk = (N + 15) / 16;

  // Degrees (with self loop) -> dinv
  k_init<<<nBlkN, 256, 0, stream>>>(dinv, pool, cnt, N);
  k_deg<<<nBlkE, 256, 0, stream>>>(dst, dinv, E);
  k_rsqrt<<<nBlkN, 256, 0, stream>>>(dinv, N);

  // Layer 1: xw1 = x@W1 ; agg1 = selfloop + edge scatter ; h1 = relu(agg1+b1)
  k_gemm128<<<gemmBlk, 256, 0, stream>>>(x, W1, bufA, N);
  k_selfloop<<<nBlkN32, 256, 0, stream>>>(bufA, dinv, bufB, N);
  k_edge<<<nBlkEw, 256, 0, stream>>>(bufA, bufB, src, dst, dinv, E);
  k_bias_relu<<<nBlkN32, 256, 0, stream>>>(bufB, b1, N);

  // Layer 2 (ping-pong: h1 in bufB -> xw2 in bufA -> agg2/h2 in bufB)
  k_gemm128<<<gemmBlk, 256, 0, stream>>>(bufB, W2, bufA, N);
  k_selfloop<<<nBlkN32, 256, 0, stream>>>(bufA, dinv, bufB, N);
  k_edge<<<nBlkEw, 256, 0, stream>>>(bufA, bufB, src, dst, dinv, E);
  k_bias_relu<<<nBlkN32, 256, 0, stream>>>(bufB, b2, N);

  // Pool + head
  k_pool<<<nBlkNw, 256, 0, stream>>>(bufB, bt, pool, cnt, N);
  k_final<<<1, 512, 0, stream>>>(pool, cnt, Wl, bl, out);
}